// Append_func_2989297238461
// MI455X (gfx1250) — compile-verified
//
#include <hip/hip_runtime.h>
#include <hip/hip_bf16.h>

#define N_NODES 100000
#define DIMF    48
#define N_EDGES 1600000
// out = z - COEFF * grad, grad = (2/N) * nf * acc  ->  scale = 0.1 * 2 / N
#define GRAD_SCALE (0.1f * 2.0f / (float)N_NODES)

typedef float v4f __attribute__((ext_vector_type(4)));

// ---------------------------------------------------------------------------
// Kernel: per-edge scatter (placed FIRST in file so the asm snippet shows it).
// 12 threads/edge (one float4 of D=48 each), 16 edges per 192-thread block.
// acc[row] += (zr[row]-zr[col]), acc[col] -= (zr[row]-zr[col]) via
// non-returning f32 atomics; acc (19.2 MB) stays resident in the 192 MB L2.
// ---------------------------------------------------------------------------
__device__ __forceinline__ void atom_add_f32(float* p, float v) {
    (void)__hip_atomic_fetch_add(p, v, __ATOMIC_RELAXED, __HIP_MEMORY_SCOPE_AGENT);
}

__global__ __launch_bounds__(192) void k_edge(
    const int*   __restrict__ row,
    const int*   __restrict__ col,
    const float* __restrict__ zreg,
    float*       __restrict__ acc)
{
    const int lt     = threadIdx.x;          // 0..191
    const int eLocal = lt / 12;              // 0..15
    const int f      = (lt % 12) * 4;        // feature start: 0,4,...,44
    const int e      = blockIdx.x * 16 + eLocal;   // < 1,600,000 exactly

    // Stream the edge-index arrays ahead through L2 (gfx1250 global_prefetch_b8)
    if (lt == 0) {
        int pe = e + 16 * 64;
        if (pe < N_EDGES) {
            __builtin_prefetch(&row[pe], 0, 1);
            __builtin_prefetch(&col[pe], 0, 1);
        }
    }

    // Index stream is touched once -> non-temporal load (TH=NT), keep L0/L2
    // capacity for the hot zreg/acc working set.
    const int r = __builtin_nontemporal_load(&row[e]);
    const int c = __builtin_nontemporal_load(&col[e]);

    const v4f a = *reinterpret_cast<const v4f*>(zreg + r * DIMF + f);
    const v4f b = *reinterpret_cast<const v4f*>(zreg + c * DIMF + f);
    const v4f d = a - b;

    float* pr = acc + r * DIMF + f;
    float* pc = acc + c * DIMF + f;

    atom_add_f32(pr + 0,  d.x);
    atom_add_f32(pr + 1,  d.y);
    atom_add_f32(pr + 2,  d.z);
    atom_add_f32(pr + 3,  d.w);
    atom_add_f32(pc + 0, -d.x);
    atom_add_f32(pc + 1, -d.y);
    atom_add_f32(pc + 2, -d.z);
    atom_add_f32(pc + 3, -d.w);
}

// ---------------------------------------------------------------------------
// Kernel: zreg = nf * z  (float4 vectorized), acc = 0
// ---------------------------------------------------------------------------
__global__ __launch_bounds__(256) void k_scale_zero(
    const v4f* __restrict__ z4,
    const float* __restrict__ nf,
    v4f* __restrict__ zreg4,
    v4f* __restrict__ acc4,
    int nvec)
{
    int t = blockIdx.x * 256 + threadIdx.x;
    if (t >= nvec) return;
    int node = t / (DIMF / 4);          // 12 float4s per node row
    float s  = nf[node];
    v4f v = z4[t];
    zreg4[t] = v * s;
    acc4[t]  = (v4f)(0.f);
}

// ---------------------------------------------------------------------------
// Kernel: out = z - GRAD_SCALE * nf * acc  (vectorized, NT b128 stores:
// the output is write-once, never re-read by the GPU)
// ---------------------------------------------------------------------------
__global__ __launch_bounds__(256) void k_finish(
    const v4f* __restrict__ z4,
    const float* __restrict__ nf,
    const v4f* __restrict__ acc4,
    v4f* __restrict__ out4,
    int nvec)
{
    int t = blockIdx.x * 256 + threadIdx.x;
    if (t >= nvec) return;
    int node = t / (DIMF / 4);
    float s  = nf[node] * GRAD_SCALE;
    v4f v = z4[t];
    v4f g = acc4[t];
    v4f o = v - g * s;
    __builtin_nontemporal_store(o, &out4[t]);
}

// ---------------------------------------------------------------------------
// Launch
// ---------------------------------------------------------------------------
extern "C" void kernel_launch(void* const* d_in, const int* in_sizes, int n_in,
                              void* d_out, int out_size, void* d_ws, size_t ws_size,
                              hipStream_t stream) {
    const float* z   = (const float*)d_in[0];           // [N, 48]
    // d_in[1] = x (unused in the Lap branch)
    const int*   ei  = (const int*)d_in[2];             // [2, E]: row = ei, col = ei + E
    const float* nf  = (const float*)d_in[3];           // [N, 1]

    const int* row = ei;
    const int* col = ei + N_EDGES;

    float* zreg = (float*)d_ws;                         // 19.2 MB
    float* acc  = zreg + (size_t)N_NODES * DIMF;        // 19.2 MB

    const int nvec = N_NODES * DIMF / 4;                // 1,200,000 float4s

    k_scale_zero<<<(nvec + 255) / 256, 256, 0, stream>>>(
        (const v4f*)z, nf, (v4f*)zreg, (v4f*)acc, nvec);

    k_edge<<<N_EDGES / 16, 192, 0, stream>>>(row, col, zreg, acc);

    k_finish<<<(nvec + 255) / 256, 256, 0, stream>>>(
        (const v4f*)z, nf, (const v4f*)acc, (v4f*)d_out, nvec);
}